// Kernel_59734405153420
// MI455X (gfx1250) — compile-verified
//
#include <hip/hip_runtime.h>
#include <math.h>

#define NPATH 4864
#define NPTS  4096
#define ODIM  144

typedef float v2f __attribute__((ext_vector_type(2)));
typedef float v4f __attribute__((ext_vector_type(4)));
typedef float v8f __attribute__((ext_vector_type(8)));

// ---------------- pair metadata (lo-major, li-minor) ----------------
__device__ const int g_lo[9]  = {0,0,0,1,1,1,2,2,2};
__device__ const int g_li[9]  = {0,1,2,0,1,2,0,1,2};
__device__ const int g_len[9] = {1,1,1,1,3,3,1,3,5};
__device__ const int g_l0[9]  = {0,1,2,1,0,1,2,1,0};
__device__ const int g_beg[9] = {0,256,512,768,1024,1792,2560,2816,3584};
__device__ const int g_rb[3]  = {0,16,64};       // row/col base per l
__device__ const int g_ns[3]  = {48,112,144};    // num_summed per lo
__device__ const int g_cgoff[9][5] = {
  {0,0,0,0,0},{1,0,0,0,0},{10,0,0,0,0},{35,0,0,0,0},
  {44,53,80,0,0},{125,170,245,0,0},{350,0,0,0,0},
  {375,420,495,0,0},{600,625,700,825,1000}};

// ---------------- CG table generation (runs once per launch) ----------------
struct cx { double re, im; };
__device__ inline cx cmul(cx a, cx b){ return {a.re*b.re-a.im*b.im, a.re*b.im+a.im*b.re}; }
__device__ inline cx cconj(cx a){ return {a.re,-a.im}; }
__device__ inline cx cadd(cx a, cx b){ return {a.re+b.re, a.im+b.im}; }
__device__ inline cx cscale(cx a, double s){ return {a.re*s, a.im*s}; }

__device__ double dfact(int n){ double r=1.0; for(int i=2;i<=n;i++) r*=(double)i; return r; }

__device__ double su2cg(int j1,int j2,int j3,int m1,int m2,int m3){
  if (m3 != m1+m2) return 0.0;
  int vmin = -j1+j2+m3; if (-j1+m1 > vmin) vmin = -j1+m1; if (0 > vmin) vmin = 0;
  int vmax = j2+j3+m1; if (j3-j1+j2 < vmax) vmax = j3-j1+j2; if (j3+m3 < vmax) vmax = j3+m3;
  if (vmax < vmin) return 0.0;
  double C = sqrt((2.0*j3+1.0)*dfact(j3+j1-j2)*dfact(j3-j1+j2)*dfact(j1+j2-j3)*
                  dfact(j3+m3)*dfact(j3-m3)/
                  (dfact(j1+j2+j3+1)*dfact(j1-m1)*dfact(j1+m1)*dfact(j2-m2)*dfact(j2+m2)));
  double S = 0.0;
  for (int v=vmin; v<=vmax; v++){
    double t = dfact(j2+j3+m1-v)*dfact(j1-m1+v)/
               (dfact(v)*dfact(j3-j1+j2-v)*dfact(j3+m3-v)*dfact(v+j1-j2-m3));
    S += (((v+j2+m2)&1)? -t : t);
  }
  return C*S;
}

__device__ void qmat(int l, cx q[9][9]){
  for(int a=0;a<9;a++) for(int b=0;b<9;b++) q[a][b] = {0.0,0.0};
  const double s2 = 0.70710678118654752440;
  for(int m=-l;m<0;m++){ q[l+m][l-m] = {s2,0.0}; q[l+m][l+m] = {0.0,-s2}; }
  q[l][l] = {1.0,0.0};
  for(int m=1;m<=l;m++){ double sg = (m&1)? -s2 : s2; q[l+m][l+m]={sg,0.0}; q[l+m][l-m]={0.0,sg}; }
  cx ph;
  switch(l&3){ case 0: ph={1,0}; break; case 1: ph={0,-1}; break;
               case 2: ph={-1,0}; break; default: ph={0,1}; }
  for(int a=0;a<9;a++) for(int b=0;b<9;b++) q[a][b] = cmul(ph, q[a][b]);
}

__global__ void cg_init(float* __restrict__ cg){
  const int plo[19] = {0,0,0,1,1,1,1,1,1,1,2,2,2,2,2,2,2,2,2};
  const int pli[19] = {0,1,2,0,1,1,1,2,2,2,0,1,1,1,2,2,2,2,2};
  const int pl [19] = {0,1,2,1,0,1,2,1,2,3,2,1,2,3,0,1,2,3,4};
  const int pof[19] = {0,1,10,35,44,53,80,125,170,245,350,375,420,495,600,625,700,825,1000};
  int t = threadIdx.x;
  if (t >= 19) return;
  int lo = plo[t], li = pli[t], l = pl[t], off = pof[t];
  cx q1[9][9], q2[9][9], q3[9][9];
  qmat(lo,q1); qmat(li,q2); qmat(l,q3);
  int n1 = 2*lo+1, n2 = 2*li+1, n3 = 2*l+1;
  double buf[225];
  double nrm = 0.0;
  for (int a=0;a<n1;a++) for (int b=0;b<n2;b++) for (int c=0;c<n3;c++){
    cx s = {0.0,0.0};
    for (int i=0;i<n1;i++) for (int k=0;k<n2;k++){
      int n = (i-lo)+(k-li)+l;
      if (n < 0 || n >= n3) continue;
      double Cv = su2cg(lo,li,l, i-lo, k-li, (i-lo)+(k-li));
      if (Cv == 0.0) continue;
      s = cadd(s, cscale(cmul(cmul(q1[i][a], q2[k][b]), cconj(q3[n][c])), Cv));
    }
    buf[(a*n2+b)*n3+c] = s.re;
    nrm += s.re*s.re;
  }
  double inv = (nrm > 0.0) ? 1.0/sqrt(nrm) : 1.0;
  for (int e=0; e<n1*n2*n3; e++) cg[off+e] = (float)(buf[e]*inv);
}

// ---------------- fused main kernel ----------------
__global__ __launch_bounds__(256) void fused_tp(
    const float* __restrict__ r, const float* __restrict__ W,
    const float* __restrict__ cg, float* __restrict__ out){
  __shared__ float sY[16][25];
  __shared__ float sB[16][32];
  __shared__ float sC[16][384];      // coeff tile (max 384 cols/chunk)
  __shared__ float sCY[16*125];      // CY[z][lidx][i][j], max 5*5*5 per point
  const int tid  = threadIdx.x;
  const int zbase = blockIdx.x * 16;

  // ---- Stage A: Y + radial basis per point ----
  if (tid < 16){
    int idx = zbase + tid;
    float x = r[idx*3+0], y = r[idx*3+1], zz = r[idx*3+2];
    float rad = sqrtf(x*x + y*y + zz*zz);
    float invr = (rad > 0.f) ? 1.f/rad : 0.f;
    float ux = x*invr, uy = y*invr, uz = zz*invr;
    float Aa[5], Bb[5];
    Aa[0] = 1.f; Bb[0] = 0.f;
    for (int m=1; m<=4; m++){
      Aa[m] = ux*Aa[m-1] - uy*Bb[m-1];
      Bb[m] = ux*Bb[m-1] + uy*Aa[m-1];
    }
    float Q[5][5];
    float dd = 1.f;
    for (int m=0; m<=4; m++){
      Q[m][m] = dd;
      if (m+1 <= 4) Q[m+1][m] = (2*m+1)*uz*Q[m][m];
      dd *= (float)(2*m+1);
    }
    for (int m=0; m<=4; m++)
      for (int l=m+2; l<=4; l++)
        Q[l][m] = ((2*l-1)*uz*Q[l-1][m] - (l-1+m)*Q[l-2][m]) / (float)(l-m);
    const float fourpi = 12.566370614359172f;
    int row = 0;
    for (int l=0; l<=4; l++){
      for (int m=-l; m<=l; m++){
        int am = (m<0)? -m : m;
        float fr = 1.f;
        for (int q=l-am+1; q<=l+am; q++) fr *= (float)q;     // (l+am)!/(l-am)!
        float N = sqrtf((float)(2*l+1)/(fourpi*fr));
        float val;
        if (m == 0)      val = N*Q[l][0];
        else if (m > 0)  val = 1.41421356237f*N*Q[l][am]*Aa[am];
        else             val = 1.41421356237f*N*Q[l][am]*Bb[am];
        sY[tid][row++] = val;
      }
    }
    for (int i=0; i<32; i++){
      float cen = 3.0f*(float)i/31.0f;
      float d = (32.0f/3.0f)*(rad - cen);
      sB[tid][i] = __expf(-d*d);
    }
  }
  __syncthreads();

  // ---- A-matrix fragments (basis tile, 16x32 f32, shared by all pairs) ----
  const int lane = tid & 31;
  const int wave = tid >> 5;
  const int mrow = lane & 15;
  const int half = lane >> 4;
  v2f afrag[8];
#pragma unroll
  for (int s=0; s<8; s++){
    afrag[s].x = sB[mrow][4*s + 2*half + 0];
    afrag[s].y = sB[mrow][4*s + 2*half + 1];
  }

#pragma unroll
  for (int p=0; p<9; p++){
    const int lo = g_lo[p], li = g_li[p], len = g_len[p], l0 = g_l0[p];
    const int n1 = 2*lo+1, n2 = 2*li+1;
    const int cbase = g_beg[p];
    // CY[z][lidx][i][j] = sum_k CG[i,j,k] * Y[z, l^2+k]
    const int E = 16*len*n1*n2;
    for (int e=tid; e<E; e+=256){
      int z    = e / (len*n1*n2);
      int rem  = e - z*(len*n1*n2);
      int lidx = rem / (n1*n2);
      int r2   = rem - lidx*(n1*n2);
      int i    = r2 / n2, j = r2 - i*n2;
      int l    = l0 + lidx;
      const float* tab = cg + g_cgoff[p][lidx];
      float acc = 0.f;
      for (int k=0; k<2*l+1; k++)
        acc += tab[(i*n2+j)*(2*l+1)+k] * sY[z][l*l+k];
      sCY[e] = acc;
    }
    __syncthreads();

    const int cu    = (len==1) ? 16 : ((len==3) ? 8 : 4);  // u's per chunk
    const int nch   = 16 / cu;
    const int ncols = cu*16*len;
    const int ntile = ncols >> 4;
    const float xf  = sqrtf((float)(2*li+1)*12.566370614359172f/(float)g_ns[lo]);

    for (int ch=0; ch<nch; ch++){
      const int colbase = cbase + ch*ncols;
      // ---- coeff tile via v_wmma_f32_16x16x4_f32, K=32 in 8 steps ----
      for (int t=wave; t<ntile; t+=8){
        v8f acc = {0.f,0.f,0.f,0.f,0.f,0.f,0.f,0.f};
        const int col = colbase + t*16 + mrow;
#pragma unroll
        for (int s=0; s<8; s++){
          v2f b;
          b.x = W[(4*s + 2*half + 0)*NPATH + col];
          b.y = W[(4*s + 2*half + 1)*NPATH + col];
          acc = __builtin_amdgcn_wmma_f32_16x16x4_f32(
                  false, afrag[s], false, b, (short)0, acc, false, false);
        }
#pragma unroll
        for (int v=0; v<8; v++)
          sC[v + 8*half][t*16 + mrow] = acc[v];
      }
      __syncthreads();

      // ---- expansion + vectorized streaming stores (b128, non-temporal) ----
      const int rowspan = cu*n1;
      const int cspan   = 16*n2;         // always a multiple of 4
      const int ngr     = cspan >> 2;    // float4 groups per output row
      const int tot4    = 16*rowspan*ngr;
      const int u0      = ch*cu;
      for (int e=tid; e<tot4; e+=256){
        int z   = e / (rowspan*ngr);
        int rem = e - z*(rowspan*ngr);
        int rl  = rem / ngr;
        int cg4 = rem - rl*ngr;
        int ul  = rl / n1, i = rl - ul*n1;
        const float* cyb = &sCY[(z*len*n1 + i)*n2];
        v4f o;
#pragma unroll
        for (int q=0; q<4; q++){
          int cl = cg4*4 + q;
          int v  = cl / n2, j = cl - v*n2;
          float acc = 0.f;
          const float* czp = &sC[z][(ul*16+v)*len];
          for (int k=0; k<len; k++)
            acc += cyb[k*n1*n2 + j] * czp[k];
          o[q] = xf*acc;
        }
        int orow = g_rb[lo] + (u0+ul)*n1 + i;
        int ocol = g_rb[li] + cg4*4;
        v4f* dst = (v4f*)&out[(size_t)(zbase+z)*(ODIM*ODIM)
                              + (size_t)orow*ODIM + ocol];
        __builtin_nontemporal_store(o, dst);
      }
      __syncthreads();
    }
  }
}

extern "C" void kernel_launch(void* const* d_in, const int* in_sizes, int n_in,
                              void* d_out, int out_size, void* d_ws, size_t ws_size,
                              hipStream_t stream){
  const float* r = (const float*)d_in[0];
  const float* W = (const float*)d_in[1];
  float* out = (float*)d_out;
  float* cg  = (float*)d_ws;          // 1225 floats of CG tables
  cg_init<<<1, 32, 0, stream>>>(cg);
  fused_tp<<<NPTS/16, 256, 0, stream>>>(r, W, cg, out);
}